// ScaledDotProductAttention_38628935860911
// MI455X (gfx1250) — compile-verified
//
#include <hip/hip_runtime.h>
#include <hip/hip_bf16.h>

// CDNA5 (gfx1250) wave32 WMMA attention: one workgroup (4 waves) per batch row.
typedef __attribute__((ext_vector_type(2))) float v2f;
typedef __attribute__((ext_vector_type(8))) float v8f;

#define SDPA_T 1024
#define SDPA_D 64
#define SDPA_BLOCK 128
#define TEMP_INV 0.125f
#define MASK_FILL -1000.0f

__global__ __launch_bounds__(SDPA_BLOCK) void sdpa_wmma_kernel(
    const float* __restrict__ q,          // [B, 64]
    const float* __restrict__ k,          // [B, 1024, 64]
    const float* __restrict__ v,          // [B, 1024, 64]
    const unsigned char* __restrict__ pm, // [B, 1024] (bool, 1 byte)
    float* __restrict__ out,              // [B, 64]
    float* __restrict__ attn)             // [B, 1024]
{
    const int b    = blockIdx.x;
    const int tid  = threadIdx.x;
    const int lane = tid & 31;
    const int wave = tid >> 5;            // 0..3

    __shared__ float sQ[SDPA_D];
    __shared__ float sS[SDPA_T];
    __shared__ float sRed[SDPA_BLOCK];
    __shared__ float sOut[SDPA_D];

    const float* Kb = k + (size_t)b * SDPA_T * SDPA_D;
    const float* Vb = v + (size_t)b * SDPA_T * SDPA_D;

    if (tid < SDPA_D) {
        sQ[tid]   = q[(size_t)b * SDPA_D + tid];
        sOut[tid] = 0.0f;
    }
    __syncthreads();

    // ---- WMMA B operand: q replicated across all 16 columns.
    // f32 16x16x4 B layout (mirrors C/D striping): VGPR0 holds row k=(lane<16?0:2),
    // VGPR1 holds row k=(lane<16?1:3). Columns identical since q is broadcast.
    const int koff = (lane < 16) ? 0 : 2;
    v2f qreg[16];
#pragma unroll
    for (int j = 0; j < 16; ++j) {
        qreg[j].x = sQ[4 * j + koff];
        qreg[j].y = sQ[4 * j + koff + 1];
    }

    // ================= Phase 1: s[t] = q . K[t,:]  (V_WMMA_F32_16X16X4_F32) ====
    // A (16x4 f32): lane m=lane&15 holds K[t0+m, d0+koff .. +koff+1] in 2 VGPRs.
    const int row = lane & 15;
    const int tilesPerWave = SDPA_T / (16 * 4);   // 16 tiles of 16 t's per wave
    for (int tt = 0; tt < tilesPerWave; ++tt) {
        const int t0 = (wave * tilesPerWave + tt) * 16;
        const float* krow = Kb + (size_t)(t0 + row) * SDPA_D + koff;
        // prefetch next K tile (speculative; dropped silently if OOB)
        __builtin_prefetch(krow + 16 * SDPA_D, 0, 0);

        v8f acc = {0.f, 0.f, 0.f, 0.f, 0.f, 0.f, 0.f, 0.f};
#pragma unroll
        for (int j = 0; j < 16; ++j) {
            v2f a = __builtin_nontemporal_load((const v2f*)(krow + 4 * j));
            acc = __builtin_amdgcn_wmma_f32_16x16x4_f32(
                false, a, false, qreg[j], (short)0, acc, false, false);
        }
        // D tile: VGPR r, lanes0-15 -> (M=r), lanes16-31 -> (M=8+r); all columns
        // identical, so lanes 0 and 16 each store 8 contiguous s-values.
        if ((lane & 15) == 0) {
            const int tb = t0 + ((lane < 16) ? 0 : 8);
#pragma unroll
            for (int r = 0; r < 8; ++r) sS[tb + r] = acc[r];
        }
    }
    __syncthreads();

    // ================= Phase 2: masked softmax over T =========================
    const unsigned char* mb = pm + (size_t)b * SDPA_T;
    float loc[SDPA_T / SDPA_BLOCK];               // 8 per thread
    float m = -3.0e38f;
#pragma unroll
    for (int i = 0; i < 8; ++i) {
        const int t = tid + i * SDPA_BLOCK;
        float s = sS[t] * TEMP_INV;
        s = mb[t] ? MASK_FILL : s;
        loc[i] = s;
        m = fmaxf(m, s);
    }
    sRed[tid] = m;
    __syncthreads();
    for (int off = SDPA_BLOCK / 2; off > 0; off >>= 1) {
        if (tid < off) sRed[tid] = fmaxf(sRed[tid], sRed[tid + off]);
        __syncthreads();
    }
    m = sRed[0];
    __syncthreads();

    float lsum = 0.f;
#pragma unroll
    for (int i = 0; i < 8; ++i) {
        const float e = __expf(loc[i] - m);
        loc[i] = e;
        lsum += e;
    }
    sRed[tid] = lsum;
    __syncthreads();
    for (int off = SDPA_BLOCK / 2; off > 0; off >>= 1) {
        if (tid < off) sRed[tid] += sRed[tid + off];
        __syncthreads();
    }
    const float inv = 1.0f / sRed[0];
    float* attnb = attn + (size_t)b * SDPA_T;
#pragma unroll
    for (int i = 0; i < 8; ++i) {
        const int t = tid + i * SDPA_BLOCK;
        const float p = loc[i] * inv;
        sS[t] = p;                                  // keep for phase 3
        __builtin_nontemporal_store(p, attnb + t);  // coalesced attn output
    }
    __syncthreads();

    // ================= Phase 3: out[d] = sum_t attn[t] * V[t,d] ===============
    // Lane l owns d = 2l, 2l+1; one wave instruction reads a full 256B V row.
    float a0 = 0.f, a1 = 0.f;
    const float* vbase = Vb + 2 * lane;
#pragma unroll 4
    for (int i = 0; i < SDPA_T / 4; ++i) {
        const int t = wave * (SDPA_T / 4) + i;
        const float p = sS[t];                      // LDS broadcast read
        v2f vv = __builtin_nontemporal_load((const v2f*)(vbase + (size_t)t * SDPA_D));
        a0 = fmaf(p, vv.x, a0);
        a1 = fmaf(p, vv.y, a1);
    }
    atomicAdd(&sOut[2 * lane], a0);                 // ds_add_f32 cross-wave reduce
    atomicAdd(&sOut[2 * lane + 1], a1);
    __syncthreads();
    if (tid < SDPA_D) out[(size_t)b * SDPA_D + tid] = sOut[tid];
}

extern "C" void kernel_launch(void* const* d_in, const int* in_sizes, int n_in,
                              void* d_out, int out_size, void* d_ws, size_t ws_size,
                              hipStream_t stream) {
    (void)n_in; (void)out_size; (void)d_ws; (void)ws_size;
    const float* q = (const float*)d_in[0];
    const float* k = (const float*)d_in[1];
    const float* v = (const float*)d_in[2];
    const unsigned char* pm = (const unsigned char*)d_in[3]; // jnp.bool_ -> 1B

    const int B = in_sizes[0] / SDPA_D;             // q is [B, 64]
    float* out  = (float*)d_out;                    // [B, 1, 64] first
    float* attn = out + (size_t)B * SDPA_D;         // then [B, 1, 1024]

    sdpa_wmma_kernel<<<B, SDPA_BLOCK, 0, stream>>>(q, k, v, pm, out, attn);
}